// PointPillar_5961414607101
// MI455X (gfx1250) — compile-verified
//
#include <hip/hip_runtime.h>
#include <cstdint>
#include <cstddef>

namespace {

constexpr int   kB      = 4;
constexpr int   kN      = 321408;
constexpr int   kPre    = 4096;
constexpr int   kPost   = 500;
constexpr int   kBins   = 2048;
constexpr int   kCand   = 8192;          // power of two, >= worst-case candidates past cutoff bin
constexpr int   kColW   = kPre / 32;     // 128 mask words per row
constexpr int   kRowT   = kPre / 16;     // 256 row tiles
constexpr float kScoreTh = 0.1f;
constexpr float kIouTh   = 0.1f;

typedef float    v2f __attribute__((ext_vector_type(2)));
typedef float    v8f __attribute__((ext_vector_type(8)));
typedef uint32_t v4u __attribute__((ext_vector_type(4)));
typedef uint32_t v8u __attribute__((ext_vector_type(8)));

// ---------------------------------------------------------------- init scratch
__global__ void k_init(uint32_t* __restrict__ hist, int* __restrict__ cnt,
                       uint32_t* __restrict__ keep) {
  const int tid = (int)threadIdx.x;
  for (int j = tid; j < kB * kBins; j += (int)blockDim.x) hist[j] = 0u;
  for (int j = tid; j < kB;         j += (int)blockDim.x) cnt[j]  = 0;
  for (int j = tid; j < kB * kColW; j += (int)blockDim.x) keep[j] = 0u;
}

// --------------------------------------------- scores = max over classes + hist
__global__ __launch_bounds__(256) void k_score_hist(const float* __restrict__ cls,
                                                    float* __restrict__ scores,
                                                    uint32_t* __restrict__ hist) {
  __shared__ uint32_t h[kBins];
  for (int i = (int)threadIdx.x; i < kBins; i += (int)blockDim.x) h[i] = 0u;
  __syncthreads();
  const int b = (int)blockIdx.y;
  const int n = (int)(blockIdx.x * blockDim.x + threadIdx.x);
  if (n < kN) {
    const float* p = cls + ((size_t)b * kN + n) * 3;
    __builtin_prefetch(p + 768, 0, 0);   // global_prefetch_b8 for the next block's lines
    const float s = fmaxf(p[0], fmaxf(p[1], p[2]));
    scores[(size_t)b * kN + n] = s;
    if (s > kScoreTh) {
      int bin = (int)(s * (float)kBins);
      bin = bin < 0 ? 0 : (bin > kBins - 1 ? kBins - 1 : bin);
      atomicAdd(&h[bin], 1u);
    }
  }
  __syncthreads();
  for (int i = (int)threadIdx.x; i < kBins; i += (int)blockDim.x)
    if (h[i]) atomicAdd(&hist[(size_t)b * kBins + i], h[i]);
}

// ------------------------------------------------- find cutoff bin (suffix>=4096)
__global__ void k_cutoff(const uint32_t* __restrict__ hist, int* __restrict__ cut) {
  if (threadIdx.x == 0) {
    const int b = (int)blockIdx.x;
    uint32_t acc = 0;
    int t = 0;
    for (int i = kBins - 1; i >= 0; --i) {
      acc += hist[(size_t)b * kBins + i];
      if (acc >= (uint32_t)kPre) { t = i; break; }
    }
    cut[b] = t;
  }
}

// ---------------------------------------------------------------- compact cands
__global__ __launch_bounds__(256) void k_compact(const float* __restrict__ scores,
                                                 const int* __restrict__ cut,
                                                 int* __restrict__ cnt,
                                                 uint64_t* __restrict__ cand) {
  const int b = (int)blockIdx.y;
  const int n = (int)(blockIdx.x * blockDim.x + threadIdx.x);
  if (n >= kN) return;
  const float s = scores[(size_t)b * kN + n];
  if (s > kScoreTh) {
    int bin = (int)(s * (float)kBins);
    bin = bin < 0 ? 0 : (bin > kBins - 1 ? kBins - 1 : bin);
    if (bin >= cut[b]) {
      const int pos = atomicAdd(&cnt[b], 1);
      if (pos < kCand) {
        const uint32_t bits = __float_as_uint(s);
        const uint32_t mk   = (bits >> 31) ? ~bits : (bits | 0x80000000u); // order-preserving
        cand[(size_t)b * kCand + pos] =
            ((uint64_t)mk << 32) | (uint64_t)(0xFFFFFFFFu - (uint32_t)n);  // ties: lower idx first
      }
    }
  }
}

// -------------------------- sort 8192 candidates desc, gather top-4096 features
__global__ __launch_bounds__(1024) void k_sort_gather(
    const float* __restrict__ cls, const float* __restrict__ boxes,
    const uint64_t* __restrict__ cand, const int* __restrict__ candCnt,
    uint32_t* __restrict__ topIdx, float* __restrict__ topScore,
    int* __restrict__ topLabel,
    float* __restrict__ fx1, float* __restrict__ fy1,
    float* __restrict__ fx2, float* __restrict__ fy2, float* __restrict__ fa) {
  __shared__ uint64_t sc[kCand];   // 64 KB of the 320 KB WGP LDS
  const int b = (int)blockIdx.x;
  int cnt = candCnt[b];
  cnt = cnt > kCand ? kCand : cnt;
  const uint64_t* src = cand + (size_t)b * kCand;

  // TDM: DMA the whole candidate array into LDS (D# group0 + group1, 1-D tile of
  // 8192 x 8B). Tracked on TENSORcnt; issued by wave 0 only (uniform branch).
  if ((threadIdx.x >> 5) == 0) {
    const uint64_t ga = (uint64_t)(uintptr_t)src;
    v4u g0;
    g0[0] = 1u;                                            // count=1, user D#
    g0[1] = (uint32_t)(uintptr_t)(&sc[0]);                 // lds_addr (low 32 bits = LDS offset)
    g0[2] = (uint32_t)ga;                                  // global_addr[31:0]
    g0[3] = (uint32_t)((ga >> 32) & 0x01FFFFFFu) | 0x80000000u; // addr[56:32] | type=2
    v8u g1;
    g1[0] = 0x00030000u;                  // workgroup_mask=0, data_size=3 (8B)
    g1[1] = (uint32_t)kCand << 16;        // tensor_dim0[15:0] in bits 79:64? -> low16 @63:48
    g1[2] = 0x00010000u;                  // tensor_dim0 hi=0, tensor_dim1=1 (low16)
    g1[3] = (uint32_t)kCand << 16;        // tile_dim0 = 8192
    g1[4] = 0x00000001u;                  // tile_dim1 = 1, tile_dim2 unused
    g1[5] = (uint32_t)kCand;              // tensor_dim0_stride low 32
    g1[6] = 0x20000000u;                  // stride0 hi=0, tensor_dim1_stride low16 = 8192
    g1[7] = 0u;
    asm volatile("tensor_load_to_lds %0, %1" :: "s"(g0), "s"(g1) : "memory");
    __builtin_amdgcn_s_wait_tensorcnt(0);
  }
  __syncthreads();
  // Redundant plain fill (same data + deterministic padding of tail slots).
  for (int i = (int)threadIdx.x; i < kCand; i += (int)blockDim.x)
    sc[i] = (i < cnt) ? src[i] : 0ull;
  __syncthreads();

  // Bitonic sort, descending (zeros/padding sink to the end).
  for (unsigned k = 2; k <= (unsigned)kCand; k <<= 1) {
    for (unsigned j = k >> 1; j > 0; j >>= 1) {
      for (unsigned e = threadIdx.x; e < (unsigned)kCand; e += blockDim.x) {
        const unsigned x = e ^ j;
        if (x > e) {
          const uint64_t a = sc[e], c = sc[x];
          const bool desc = ((e & k) == 0);
          if (desc ? (a < c) : (a > c)) { sc[e] = c; sc[x] = a; }
        }
      }
      __syncthreads();
    }
  }

  // Emit top-4096: score, original index, label, BEV box features.
  for (int i = (int)threadIdx.x; i < kPre; i += (int)blockDim.x) {
    const uint64_t v = sc[i];
    uint32_t idx = 0xFFFFFFFFu - (uint32_t)(v & 0xFFFFFFFFull);
    float s;
    if (v == 0ull) { s = -1.0f; idx = 0xFFFFFFFFu; }
    else {
      const uint32_t mk   = (uint32_t)(v >> 32);
      const uint32_t bits = (mk & 0x80000000u) ? (mk ^ 0x80000000u) : ~mk;
      s = __uint_as_float(bits);
    }
    const size_t o = (size_t)b * kPre + i;
    topScore[o] = s;
    topIdx[o]   = idx;
    if (idx < (uint32_t)kN) {
      const float* bp = boxes + ((size_t)b * kN + idx) * 7;
      const float x = bp[0], y = bp[1], dx = bp[3], dy = bp[4];
      fx1[o] = x - 0.5f * dx; fx2[o] = x + 0.5f * dx;
      fy1[o] = y - 0.5f * dy; fy2[o] = y + 0.5f * dy;
      fa[o]  = dx * dy;
      const float* cp = cls + ((size_t)b * kN + idx) * 3;
      int lab = 1; float bb = cp[0];
      if (cp[1] > bb) { bb = cp[1]; lab = 2; }
      if (cp[2] > bb) { lab = 3; }
      topLabel[o] = lab;
    } else {
      fx1[o] = 0.f; fy1[o] = 0.f; fx2[o] = 0.f; fy2[o] = 0.f; fa[o] = 0.f;
      topLabel[o] = 0;
    }
  }
}

// ------------- suppression mask: one wave per 16x32 tile, WMMA for area outer-sum
__global__ __launch_bounds__(256) void k_mask(const float* __restrict__ fx1,
                                              const float* __restrict__ fy1,
                                              const float* __restrict__ fx2,
                                              const float* __restrict__ fy2,
                                              const float* __restrict__ fa,
                                              uint32_t* __restrict__ mask) {
  const int wave = (int)((blockIdx.x * blockDim.x + threadIdx.x) >> 5);
  const int lane = (int)(threadIdx.x & 31);
  const int tilesPerBatch = kRowT * kColW;       // 256 * 128
  const int b  = wave / tilesPerBatch;
  const int t  = wave % tilesPerBatch;
  const int rT = t / kColW;
  const int cT = t % kColW;
  const int base = b * kPre;

  const int rowIdx = rT * 16 + (lane & 15);      // lanes 0-15 / 16-31 both load rows 0..15
  const float rx1 = fx1[base + rowIdx], ry1 = fy1[base + rowIdx];
  const float rx2 = fx2[base + rowIdx], ry2 = fy2[base + rowIdx];
  const float rar = fa [base + rowIdx];

  const int c0 = cT * 32 + (lane & 15);
  const int c1 = c0 + 16;
  const float ax1 = fx1[base + c0], ay1 = fy1[base + c0];
  const float ax2 = fx2[base + c0], ay2 = fy2[base + c0];
  const float aar = fa [base + c0];
  const float bx1 = fx1[base + c1], by1 = fy1[base + c1];
  const float bx2 = fx2[base + c1], by2 = fy2[base + c1];
  const float bar = fa [base + c1];

  // u[m][n] = area_row[m] + area_col[n] via two rank-1 WMMA accumulations each.
  // A constant across K, B constant across K -> layout-robust under documented
  // lane->M (A: lane%16) and lane->N (B/D: lane%16), D: M = vgpr + 8*(lane/16).
  const float qr = 0.25f * rar;
  v2f A1;    A1[0] = qr;    A1[1] = qr;
  v2f Bone;  Bone[0] = 1.f; Bone[1] = 1.f;
  v2f Aq;    Aq[0] = 0.25f; Aq[1] = 0.25f;
  v2f Ba;    Ba[0] = aar;   Ba[1] = aar;
  v2f Bb;    Bb[0] = bar;   Bb[1] = bar;
  v8f u0 = {};
  v8f u1 = {};
  u0 = __builtin_amdgcn_wmma_f32_16x16x4_f32(false, A1, false, Bone, (short)0, u0, false, false);
  u0 = __builtin_amdgcn_wmma_f32_16x16x4_f32(false, Aq, false, Ba,   (short)0, u0, false, false);
  u1 = __builtin_amdgcn_wmma_f32_16x16x4_f32(false, A1, false, Bone, (short)0, u1, false, false);
  u1 = __builtin_amdgcn_wmma_f32_16x16x4_f32(false, Aq, false, Bb,   (short)0, u1, false, false);

  const int hi = lane >> 4;
#pragma unroll
  for (int v = 0; v < 8; ++v) {
    const int src = v + 8 * hi;                  // row M for this lane's accumulator elem
    const float mx1 = __shfl(rx1, src, 32);
    const float my1 = __shfl(ry1, src, 32);
    const float mx2 = __shfl(rx2, src, 32);
    const float my2 = __shfl(ry2, src, 32);

    float iw = fmaxf(fminf(mx2, ax2) - fmaxf(mx1, ax1), 0.0f);
    float ih = fmaxf(fminf(my2, ay2) - fmaxf(my1, ay1), 0.0f);
    float inter = iw * ih;
    float iou = inter / fmaxf(u0[v] - inter, 1e-6f);
    const bool p0 = iou > kIouTh;

    iw = fmaxf(fminf(mx2, bx2) - fmaxf(mx1, bx1), 0.0f);
    ih = fmaxf(fminf(my2, by2) - fmaxf(my1, by1), 0.0f);
    inter = iw * ih;
    iou = inter / fmaxf(u1[v] - inter, 1e-6f);
    const bool p1 = iou > kIouTh;

    const uint32_t w0 = __builtin_amdgcn_ballot_w32(p0);
    const uint32_t w1 = __builtin_amdgcn_ballot_w32(p1);
    const uint32_t lo  = (w0 & 0xFFFFu) | ((w1 & 0xFFFFu) << 16);
    const uint32_t hiw = (w0 >> 16)     | (w1 & 0xFFFF0000u);
    const int rowLo = rT * 16 + v;
    if (lane == 0)  mask[((size_t)base + rowLo)     * kColW + cT] = lo;
    if (lane == 16) mask[((size_t)base + rowLo + 8) * kColW + cT] = hiw;
  }
}

// -------- greedy NMS: single wave per batch; DS ops are in-order per wave32
__global__ __launch_bounds__(32) void k_nms(const float* __restrict__ topScore,
                                            const uint32_t* __restrict__ mask,
                                            uint32_t* __restrict__ keepW) {
  __shared__ float    ssc[kPre];
  __shared__ uint32_t removed[kColW];
  const int b = (int)blockIdx.x;
  const int lane = (int)threadIdx.x;
  for (int j = lane; j < kPre; j += 32) ssc[j] = topScore[(size_t)b * kPre + j];
#pragma unroll
  for (int t = 0; t < kColW / 32; ++t) removed[lane + 32 * t] = 0u;
  __syncthreads();
  volatile uint32_t* rem = removed;
  for (int i = 0; i < kPre; ++i) {
    const uint32_t rw = rem[i >> 5];
    const bool kept = (ssc[i] > kScoreTh) && !((rw >> (i & 31)) & 1u);
    if (kept) {
      const uint32_t* mrow = mask + ((size_t)b * kPre + i) * kColW;
#pragma unroll
      for (int t = 0; t < kColW / 32; ++t) {
        const int w = lane + 32 * t;
        rem[w] |= mrow[w];
      }
      if (lane == 0) keepW[b * kColW + (i >> 5)] |= (1u << (i & 31));
    }
  }
}

// ---------------------------------------------------------------- final output
__device__ inline void write_slot(float* __restrict__ out, const float* __restrict__ boxes,
                                  int b, int pos, float s, uint32_t idx, int lab) {
  float* oBox = out;
  float* oSc  = out + (size_t)kB * kPost * 7;
  float* oLb  = oSc + (size_t)kB * kPost;
  float* ob = oBox + ((size_t)b * kPost + pos) * 7;
  if (idx < (uint32_t)kN) {
    const float* bp = boxes + ((size_t)b * kN + idx) * 7;
#pragma unroll
    for (int k = 0; k < 7; ++k) ob[k] = bp[k];
  } else {
#pragma unroll
    for (int k = 0; k < 7; ++k) ob[k] = 0.0f;
  }
  oSc[(size_t)b * kPost + pos] = s;
  oLb[(size_t)b * kPost + pos] = (float)lab;
}

__global__ __launch_bounds__(32) void k_out(const float* __restrict__ boxes,
                                            const uint32_t* __restrict__ keepW,
                                            const float* __restrict__ topScore,
                                            const uint32_t* __restrict__ topIdx,
                                            const int* __restrict__ topLabel,
                                            float* __restrict__ out) {
  const int b = (int)blockIdx.x;
  const int lane = (int)threadIdx.x;
  const uint32_t lt = (1u << lane) - 1u;
  int kbase = 0;
  for (int c = 0; c < kColW; ++c) {
    const uint32_t kw = keepW[b * kColW + c];
    const int i = c * 32 + lane;
    const bool kept = (kw >> lane) & 1u;
    const int rank = kbase + __popc(kw & lt);
    if (kept && rank < kPost) {
      const size_t o = (size_t)b * kPre + i;
      write_slot(out, boxes, b, rank, topScore[o], topIdx[o], topLabel[o]);
    }
    kbase += __popc(kw);
  }
  if (kbase < kPost) {   // pad like top_k over (-1)-masked scores: non-kept, index order
    const int K = kbase;
    int nbase = 0;
    for (int c = 0; c < kColW; ++c) {
      const uint32_t kw = keepW[b * kColW + c];
      const uint32_t nw = ~kw;
      const int i = c * 32 + lane;
      const bool nk = (nw >> lane) & 1u;
      const int pos = K + nbase + __popc(nw & lt);
      if (nk && pos < kPost) {
        const size_t o = (size_t)b * kPre + i;
        write_slot(out, boxes, b, pos, -1.0f, topIdx[o], topLabel[o]);
      }
      nbase += __popc(nw);
      if (K + nbase >= kPost) break;
    }
  }
}

}  // namespace

extern "C" void kernel_launch(void* const* d_in, const int* in_sizes, int n_in,
                              void* d_out, int out_size, void* d_ws, size_t ws_size,
                              hipStream_t stream) {
  (void)in_sizes; (void)n_in; (void)out_size; (void)ws_size;
  const float* cls   = (const float*)d_in[0];   // (B,N,3) f32
  const float* boxes = (const float*)d_in[1];   // (B,N,7) f32

  char* ws = (char*)d_ws;
  size_t off = 0;
  auto take = [&](size_t bytes) -> void* {
    void* p = ws + off;
    off += (bytes + 255) & ~(size_t)255;
    return p;
  };
  uint64_t* cand   = (uint64_t*)take((size_t)kB * kCand * 8);
  float*    scores = (float*)   take((size_t)kB * kN * 4);
  uint32_t* hist   = (uint32_t*)take((size_t)kB * kBins * 4);
  int*      cut    = (int*)     take((size_t)kB * 4);
  int*      cnt    = (int*)     take((size_t)kB * 4);
  uint32_t* tidx   = (uint32_t*)take((size_t)kB * kPre * 4);
  float*    tsc    = (float*)   take((size_t)kB * kPre * 4);
  int*      tlb    = (int*)     take((size_t)kB * kPre * 4);
  float*    fx1    = (float*)   take((size_t)kB * kPre * 4);
  float*    fy1    = (float*)   take((size_t)kB * kPre * 4);
  float*    fx2    = (float*)   take((size_t)kB * kPre * 4);
  float*    fy2    = (float*)   take((size_t)kB * kPre * 4);
  float*    fa     = (float*)   take((size_t)kB * kPre * 4);
  uint32_t* mask   = (uint32_t*)take((size_t)kB * kPre * kColW * 4);   // 8 MB
  uint32_t* keep   = (uint32_t*)take((size_t)kB * kColW * 4);

  const int nBlk = (kN + 255) / 256;   // 1256

  k_init<<<1, 256, 0, stream>>>(hist, cnt, keep);
  k_score_hist<<<dim3(nBlk, kB), 256, 0, stream>>>(cls, scores, hist);
  k_cutoff<<<kB, 32, 0, stream>>>(hist, cut);
  k_compact<<<dim3(nBlk, kB), 256, 0, stream>>>(scores, cut, cnt, cand);
  k_sort_gather<<<kB, 1024, 0, stream>>>(cls, boxes, cand, cnt, tidx, tsc, tlb,
                                         fx1, fy1, fx2, fy2, fa);
  k_mask<<<(kB * kRowT * kColW) / 8, 256, 0, stream>>>(fx1, fy1, fx2, fy2, fa, mask);
  k_nms<<<kB, 32, 0, stream>>>(tsc, mask, keep);
  k_out<<<kB, 32, 0, stream>>>(boxes, keep, tsc, tidx, tlb, (float*)d_out);
}